// HybridKernelQCNNQML_65481071410788
// MI455X (gfx1250) — compile-verified
//
#include <hip/hip_runtime.h>
#include <math.h>

typedef __attribute__((ext_vector_type(2))) float v2f;
typedef __attribute__((ext_vector_type(8))) float v8f;

#define NROWS 8192
#define DIM   64
#define HID   32

// ---------------------------------------------------------------------------
// Kernel 1: per-row feature map  f = tanh(x@W1^T + b1)@W2^T + b2  and ||x||^2.
// One thread per row; weights staged in LDS (8KB + 4KB + 256B).
// ---------------------------------------------------------------------------
__global__ __launch_bounds__(256) void qmap_kernel(
    const float* __restrict__ x,
    const float* __restrict__ W1, const float* __restrict__ b1,
    const float* __restrict__ W2, const float* __restrict__ b2,
    float* __restrict__ f, float* __restrict__ xsq)
{
    __shared__ float sW1[HID * DIM];
    __shared__ float sW2[HID * HID];
    __shared__ float sb1[HID];
    __shared__ float sb2[HID];

    const int tid = threadIdx.x;
    for (int i = tid; i < HID * DIM; i += 256) sW1[i] = W1[i];
    for (int i = tid; i < HID * HID; i += 256) sW2[i] = W2[i];
    if (tid < HID) { sb1[tid] = b1[tid]; sb2[tid] = b2[tid]; }
    __syncthreads();

    const int row = blockIdx.x * 256 + tid;

    float xr[DIM];
    float sq = 0.0f;
    #pragma unroll
    for (int d = 0; d < DIM; ++d) {
        xr[d] = x[(size_t)row * DIM + d];
        sq += xr[d] * xr[d];
    }
    xsq[row] = sq;

    float h[HID];
    #pragma unroll
    for (int j = 0; j < HID; ++j) {
        float s = sb1[j];
        #pragma unroll
        for (int d = 0; d < DIM; ++d) s += xr[d] * sW1[j * DIM + d];
        h[j] = tanhf(s);
    }
    #pragma unroll
    for (int j = 0; j < HID; ++j) {
        float s = sb2[j];
        #pragma unroll
        for (int t = 0; t < HID; ++t) s += h[t] * sW2[j * HID + t];
        f[(size_t)row * HID + j] = s;
    }
}

// ---------------------------------------------------------------------------
// Kernel 2: fused Gram kernel.
//   out = 0.5 * (fa@fb^T) + 0.5 * exp(-max(a_sq + b_sq - 2*(a@b^T), 0))
// Block = 256 threads (8 waves). Block tile = 128 rows x 32 cols.
// Wave w -> rows [w*16, w*16+16), cols [0,32): two 16x16 WMMA tiles sharing A.
// All GEMM work via v_wmma_f32_16x16x4_f32 (f32 in, f32 acc).
// ---------------------------------------------------------------------------
__global__ __launch_bounds__(256) void gram_kernel(
    const float* __restrict__ a,  const float* __restrict__ b,
    const float* __restrict__ fa, const float* __restrict__ fb,
    const float* __restrict__ asq, const float* __restrict__ bsq,
    float* __restrict__ out)
{
    const int lane = threadIdx.x & 31;
    const int wave = threadIdx.x >> 5;
    const int m    = lane & 15;   // row (A frag) / col (B frag) index within tile
    const int kh   = lane >> 4;   // K half-select: 0 -> K 0..1, 1 -> K 2..3

    const int row0 = blockIdx.y * 128 + wave * 16;
    const int col0 = blockIdx.x * 32;

    v8f acc_ab0 = {}, acc_ab1 = {}, acc_q0 = {}, acc_q1 = {};

    // ---- RBF part: a @ b^T, K = 64 -> 16 WMMA steps x 2 col tiles ----
    {
        const float* arow  = a + (size_t)(row0 + m) * DIM;
        const float* brow0 = b + (size_t)(col0 + m) * DIM;
        const float* brow1 = b + (size_t)(col0 + 16 + m) * DIM;
        #pragma unroll
        for (int k = 0; k < DIM; k += 4) {
            v2f af  = *(const v2f*)(arow  + k + 2 * kh);
            v2f bf0 = *(const v2f*)(brow0 + k + 2 * kh);
            v2f bf1 = *(const v2f*)(brow1 + k + 2 * kh);
            acc_ab0 = __builtin_amdgcn_wmma_f32_16x16x4_f32(
                false, af, false, bf0, (short)0, acc_ab0, false, false);
            acc_ab1 = __builtin_amdgcn_wmma_f32_16x16x4_f32(
                false, af, false, bf1, (short)0, acc_ab1, false, false);
        }
    }

    // ---- Quantum part: fa @ fb^T, K = 32 -> 8 WMMA steps x 2 col tiles ----
    {
        const float* farow  = fa + (size_t)(row0 + m) * HID;
        const float* fbrow0 = fb + (size_t)(col0 + m) * HID;
        const float* fbrow1 = fb + (size_t)(col0 + 16 + m) * HID;
        #pragma unroll
        for (int k = 0; k < HID; k += 4) {
            v2f af  = *(const v2f*)(farow  + k + 2 * kh);
            v2f bf0 = *(const v2f*)(fbrow0 + k + 2 * kh);
            v2f bf1 = *(const v2f*)(fbrow1 + k + 2 * kh);
            acc_q0 = __builtin_amdgcn_wmma_f32_16x16x4_f32(
                false, af, false, bf0, (short)0, acc_q0, false, false);
            acc_q1 = __builtin_amdgcn_wmma_f32_16x16x4_f32(
                false, af, false, bf1, (short)0, acc_q1, false, false);
        }
    }

    // ---- Epilogue: C layout -> VGPR r holds (row0 + r + 8*kh, col0 + m) ----
    const float bs0 = bsq[col0 + m];
    const float bs1 = bsq[col0 + 16 + m];

    #pragma unroll
    for (int r = 0; r < 8; ++r) {
        const int row = row0 + r + 8 * kh;
        const float as = asq[row];
        const float d0 = as + bs0 - 2.0f * acc_ab0[r];
        const float d1 = as + bs1 - 2.0f * acc_ab1[r];
        const float rbf0 = __expf(-fmaxf(d0, 0.0f));   // GAMMA = 1.0
        const float rbf1 = __expf(-fmaxf(d1, 0.0f));
        const float o0 = 0.5f * acc_q0[r] + 0.5f * rbf0;
        const float o1 = 0.5f * acc_q1[r] + 0.5f * rbf1;
        // one-shot 256MB output: bypass cache with non-temporal stores
        __builtin_nontemporal_store(o0, out + (size_t)row * NROWS + col0 + m);
        __builtin_nontemporal_store(o1, out + (size_t)row * NROWS + col0 + 16 + m);
    }
}

// ---------------------------------------------------------------------------
extern "C" void kernel_launch(void* const* d_in, const int* in_sizes, int n_in,
                              void* d_out, int out_size, void* d_ws, size_t ws_size,
                              hipStream_t stream) {
    const float* a  = (const float*)d_in[0];
    const float* b  = (const float*)d_in[1];
    const float* W1 = (const float*)d_in[2];
    const float* b1 = (const float*)d_in[3];
    const float* W2 = (const float*)d_in[4];
    const float* b2 = (const float*)d_in[5];
    float* out = (float*)d_out;

    // Workspace layout: fa[8192*32] | fb[8192*32] | asq[8192] | bsq[8192]
    float* fa  = (float*)d_ws;
    float* fb  = fa + (size_t)NROWS * HID;
    float* asq = fb + (size_t)NROWS * HID;
    float* bsq = asq + NROWS;

    qmap_kernel<<<NROWS / 256, 256, 0, stream>>>(a, W1, b1, W2, b2, fa, asq);
    qmap_kernel<<<NROWS / 256, 256, 0, stream>>>(b, W1, b1, W2, b2, fb, bsq);

    dim3 grid(NROWS / 32, NROWS / 128);  // (cols, rows)
    gram_kernel<<<grid, 256, 0, stream>>>(a, b, fa, fb, asq, bsq, out);
}